// Correlation_49898930045528
// MI455X (gfx1250) — compile-verified
//
#include <hip/hip_runtime.h>
#include <hip/hip_bf16.h>

typedef float v2f __attribute__((ext_vector_type(2)));
typedef float v8f __attribute__((ext_vector_type(8)));
typedef unsigned int v4u __attribute__((ext_vector_type(4)));
typedef int v8i __attribute__((ext_vector_type(8)));
typedef int v4i __attribute__((ext_vector_type(4)));

#define CCH  128      // channels (reduction dim)
#define HH   128
#define WW   128
#define KD   21       // 21x21 displacement window
#define PADR 10
#define ROWF 160      // LDS row pitch in floats: 16 apron + 128 + 16 apron
#define NTHR 256      // 8 waves; wave j owns output w-tile J=j

// Issue one TDM DMA: copy tgt row-block (128 c-rows x 128 f32, stride H*W) into
// LDS at lds_byte with a 128B pad inserted after every 512B row -> pitch 640B.
__device__ __forceinline__ void tdm_load_row(unsigned lds_byte, const float* gptr) {
    unsigned long long ga = (unsigned long long)(uintptr_t)gptr;
    v4u g0;
    g0[0] = 1u;                                   // count=1 (valid user D#)
    g0[1] = lds_byte;                             // lds_addr
    g0[2] = (unsigned)(ga & 0xffffffffu);         // global_addr[31:0]
    g0[3] = (unsigned)(ga >> 32) | (2u << 30);    // global_addr[56:32] | type=2
    v8i g1;
    g1[0] = (2 << 16)                             // data_size = 4B
          | (1 << 20)                             // pad_enable
          | (6 << 22)                             // pad_interval: 2^(6+1)=128 DW = 512B
          | (31 << 25);                           // pad_amount: 32 DW = 128B
    g1[1] = (int)(WW  << 16);                     // tensor_dim0 lo16 (=128)
    g1[2] = (int)(CCH << 16);                     // dim0 hi=0 | tensor_dim1 lo16 (=128)
    g1[3] = (int)(WW  << 16);                     // dim1 hi=0 | tile_dim0 = 128
    g1[4] = (int)CCH;                             // tile_dim1 = 128 | tile_dim2 = 0
    g1[5] = HH * WW;                              // tensor_dim0_stride lo32 = 16384
    g1[6] = 0;                                    // stride hi | tensor_dim1_stride lo
    g1[7] = 0;
    v4i g2 = {0, 0, 0, 0}, g3 = {0, 0, 0, 0};     // 2D: groups 2/3 unused
    v8i g4 = {0, 0, 0, 0, 0, 0, 0, 0};            // extra group (6-arg toolchain form)
    __builtin_amdgcn_tensor_load_to_lds(g0, g1, g2, g3, g4, 0);
}

// Band-matmul correlation: per (b,h), P = T^T S holds all dx-correlations on
// its diagonals; compute only the 3 diagonal-adjacent 16x16 WMMA tiles per
// output w-tile. T rows are DMA'd by the Tensor Data Mover, double-buffered.
__global__ __launch_bounds__(NTHR)
void corr_wmma_tdm_kernel(const float* __restrict__ in, float* __restrict__ out) {
    const int h    = blockIdx.x;
    const int b    = blockIdx.y;
    const int tid  = threadIdx.x;
    const int lane = tid & 31;     // wave32
    const int J    = tid >> 5;     // wave id == output w-tile (0..7)
    const int col  = lane & 15;
    const int hk   = lane >> 4;    // lane half -> K pair (f32 16x16x4 layout)

    __shared__ float Tbuf[2][CCH * ROWF];   // [c][wi], wi = w'+16; 2 x 80KB

    const float* src = in + (size_t)(b * 2 + 0) * CCH * HH * WW;
    const float* tgt = in + (size_t)(b * 2 + 1) * CCH * HH * WW;

    // ---- zero the permanent aprons (wi<16 and wi>=144 of every c row) ----
    for (int idx = tid; idx < CCH * 32; idx += NTHR) {       // 4096/256 = 16
        int c = idx >> 5, a = idx & 31;
        int wi = (a < 16) ? a : (a + 128);
        Tbuf[0][c * ROWF + wi] = 0.0f;
        Tbuf[1][c * ROWF + wi] = 0.0f;
    }

    // ---- dy-invariant B fragments straight from global into VGPRs ----
    v2f bfrag[32];
    const float* Bp = src + (size_t)(2 * hk) * HH * WW + (size_t)h * WW + (J * 16 + col);
    #pragma unroll
    for (int ks = 0; ks < 32; ++ks) {
        v2f t;
        t.x = Bp[(size_t)(4 * ks)     * HH * WW];
        t.y = Bp[(size_t)(4 * ks + 1) * HH * WW];
        bfrag[ks] = t;
    }

    const int dy_lo = (PADR - h) > 0 ? (PADR - h) : 0;             // hp >= 0
    const int dy_hi = (HH + PADR - h) < KD ? (HH + PADR - h) : KD; // hp < HH

    // ---- zero-fill outputs of fully padded dy rows ----
    for (int dy = 0; dy < KD; ++dy) {
        if (dy >= dy_lo && dy < dy_hi) continue;
        for (int idx = tid; idx < KD * WW; idx += NTHR) {
            int kdx = idx >> 7, w = idx & 127;
            out[(((size_t)b * (KD * KD) + dy * KD + kdx) * HH + h) * WW + w] = 0.0f;
        }
    }

    const float scale = 1.0f / (float)CCH;
    int cur = 0;

    // prologue: DMA first valid row block into buffer 0 (wave 0 only)
    if (J == 0)
        tdm_load_row((unsigned)(uintptr_t)&Tbuf[0][16],
                     tgt + (size_t)(h + dy_lo - PADR) * WW);

    for (int dy = dy_lo; dy < dy_hi; ++dy) {
        if (J == 0) {
            if (dy + 1 < dy_hi) {
                // overlap: DMA next row block into the other buffer
                tdm_load_row((unsigned)(uintptr_t)&Tbuf[cur ^ 1][16],
                             tgt + (size_t)(h + dy + 1 - PADR) * WW);
                __builtin_amdgcn_s_wait_tensorcnt(1);   // current buffer complete
            } else {
                __builtin_amdgcn_s_wait_tensorcnt(0);
            }
        }
        __syncthreads();   // publish buf[cur] (and aprons on first pass)

        // ---- 3 diagonal tiles per wave, sharing each B fragment ----
        // A[m,k]: lane m=col, k pair = 4ks+2hk+{0,1}; tile t adds 16t floats.
        const float* Ab = &Tbuf[cur][(2 * hk) * ROWF + 16 * J + col];
        v8f acc0 = {0.f,0.f,0.f,0.f,0.f,0.f,0.f,0.f};
        v8f acc1 = acc0, acc2 = acc0;
        #pragma unroll
        for (int ks = 0; ks < 32; ++ks) {
            const int o = ks * (4 * ROWF);
            v2f a0, a1, a2;
            a0.x = Ab[o];      a0.y = Ab[o + ROWF];
            a1.x = Ab[o + 16]; a1.y = Ab[o + ROWF + 16];
            a2.x = Ab[o + 32]; a2.y = Ab[o + ROWF + 32];
            acc0 = __builtin_amdgcn_wmma_f32_16x16x4_f32(false, a0, false, bfrag[ks], (short)0, acc0, false, false);
            acc1 = __builtin_amdgcn_wmma_f32_16x16x4_f32(false, a1, false, bfrag[ks], (short)0, acc1, false, false);
            acc2 = __builtin_amdgcn_wmma_f32_16x16x4_f32(false, a2, false, bfrag[ks], (short)0, acc2, false, false);
        }

        // ---- scatter the |dx|<=10 band; each (w,dx) covered by exactly one tile ----
        v8f accs[3] = {acc0, acc1, acc2};
        #pragma unroll
        for (int t = 0; t < 3; ++t) {
            #pragma unroll
            for (int r = 0; r < 8; ++r) {
                int m  = r + 8 * hk;
                int dx = (t - 1) * 16 + m - col;
                if (dx >= -PADR && dx <= PADR) {
                    int kk = dy * KD + dx + PADR;
                    out[(((size_t)b * (KD * KD) + kk) * HH + h) * WW + (J * 16 + col)]
                        = accs[t][r] * scale;
                }
            }
        }

        __syncthreads();   // all waves done reading buf[cur] before it is re-DMA'd
        cur ^= 1;
    }
}

extern "C" void kernel_launch(void* const* d_in, const int* in_sizes, int n_in,
                              void* d_out, int out_size, void* d_ws, size_t ws_size,
                              hipStream_t stream) {
    (void)in_sizes; (void)n_in; (void)d_ws; (void)ws_size; (void)out_size;
    const float* in  = (const float*)d_in[0];   // (4, 2, 128, 128, 128) f32
    float*       out = (float*)d_out;           // (4, 441, 128, 128) f32

    dim3 grid(HH, 4);          // (h, b)
    dim3 block(NTHR);          // 8 waves: one per 16-wide output w-tile
    corr_wmma_tdm_kernel<<<grid, block, 0, stream>>>(in, out);
}